// ImportanceSparsification_62491774157234
// MI455X (gfx1250) — compile-verified
//
#include <hip/hip_runtime.h>
#include <stdint.h>
#include <stddef.h>

// ============================================================================
// ImportanceSparsification for MI455X (gfx1250)
//   top-k of 1/(cost+eps)  ==  k smallest costs per batch  (monotone transform)
//   => exact 3-round radix select on float bit patterns (all costs positive),
//      then one streaming masked-write pass with NT store hints.
//   Memory-bound: ~160MB of HBM traffic => ~7us floor at 23.3 TB/s.
//   cost (64MB) is L2-resident (192MB L2), so extra select passes hit L2.
// ============================================================================

typedef float v4f __attribute__((ext_vector_type(4)));
typedef int   v4i __attribute__((vector_size(16)));   // matches builtin sig

#define NBATCH    16
#define NELEM     (1024 * 1024)     // S*T per batch
#define KSEL      209715u           // int(S*T*0.2)
#define SRC_ELEMS (16 * 1024 * 128) // B*S*D floats (source == target size)

// ---- workspace layout (u32 units) ------------------------------------------
#define HIST_BINS    2048
#define HIST_OFF     0
#define STATE_OFF    (NBATCH * HIST_BINS)        // {prefix, kRem} per batch
#define LISTCNT_OFF  (STATE_OFF + 2 * NBATCH)
#define LIST_CAP     256
#define LIST_OFF     (LISTCNT_OFF + NBATCH)
#define CUTOFF_OFF   (LIST_OFF + NBATCH * LIST_CAP)
#define WS_U32       (CUTOFF_OFF + NBATCH)       // ~148 KB total

// ---- gfx1250 async global->LDS DMA (ASYNCcnt path) --------------------------
// Probe result (round 1): param 1 of the b128 builtin is
//   'int __attribute__((vector_size(16))) __device__ *'  i.e. v4i addrspace(1)*
// so build exactly-typed AS pointers via integer casts (generic global address
// == AS1 address; generic LDS address truncates to LDS offset in low 32 bits).
typedef __attribute__((address_space(1))) v4i* g_v4i_ptr;
typedef __attribute__((address_space(3))) v4i* l_v4i_ptr;

#if defined(__AMDGCN__) && __has_builtin(__builtin_amdgcn_global_load_async_to_lds_b128)
#define HAVE_ASYNC_LDS 1
#else
#define HAVE_ASYNC_LDS 0
#endif

__device__ __forceinline__ void async_g2l_b128(const float* gsrc, v4f* lds_dst) {
#if HAVE_ASYNC_LDS
  __builtin_amdgcn_global_load_async_to_lds_b128(
      (g_v4i_ptr)(uintptr_t)gsrc,
      (l_v4i_ptr)(uint32_t)(uintptr_t)lds_dst,
      /*offset=*/0, /*cpol=*/0);
#else
  *lds_dst = *(const v4f*)gsrc;   // fallback: plain b128 load + ds_store
#endif
}

__device__ __forceinline__ void wait_async0() {
#if defined(__AMDGCN__) && __has_builtin(__builtin_amdgcn_s_wait_asynccnt)
  __builtin_amdgcn_s_wait_asynccnt(0);
#elif defined(__AMDGCN__)
  asm volatile("s_wait_asynccnt 0" ::: "memory");
#endif
  asm volatile("" ::: "memory");  // keep LDS reads below the wait
}
__device__ __forceinline__ void wait_async1() {
#if defined(__AMDGCN__) && __has_builtin(__builtin_amdgcn_s_wait_asynccnt)
  __builtin_amdgcn_s_wait_asynccnt(1);
#elif defined(__AMDGCN__)
  asm volatile("s_wait_asynccnt 1" ::: "memory");
#endif
  asm volatile("" ::: "memory");
}

// ---- state init -------------------------------------------------------------
__global__ void init_state_kernel(uint32_t* ws) {
  int b = threadIdx.x;
  if (b < NBATCH) {
    ws[STATE_OFF + 2 * b + 0] = 0u;    // prefix (threshold bits found so far)
    ws[STATE_OFF + 2 * b + 1] = KSEL;  // k remaining
  }
}

// ---- radix histogram pass ---------------------------------------------------
// grid = (64 chunks, 16 batches), 256 threads. Each block stages 16 tiles of
// 1024 floats through LDS via async DMA (double buffered: each lane reads back
// exactly the 16B it requested, so only s_wait_asynccnt is needed, no barrier)
// and builds a 2048-bin LDS histogram of the current radix digit.
#define STAGES 16
__global__ __launch_bounds__(256) void radix_hist_kernel(
    const float* __restrict__ cost, uint32_t* __restrict__ ws,
    uint32_t maskHigh, int shift, int nbins) {
  __shared__ uint32_t lhist[HIST_BINS];
  __shared__ v4f stage[2][256];

  const int b   = blockIdx.y;
  const int tid = threadIdx.x;
  const uint32_t prefix = ws[STATE_OFF + 2 * b];   // uniform scalar load

  for (int i = tid; i < HIST_BINS; i += 256) lhist[i] = 0u;
  __syncthreads();

  const float* base =
      cost + (size_t)b * NELEM + (size_t)blockIdx.x * (STAGES * 1024);

  async_g2l_b128(base + tid * 4, &stage[0][tid]);        // prologue
  #pragma unroll 4
  for (int s = 0; s < STAGES; ++s) {
    const int cur = s & 1;
    if (s + 1 < STAGES) {
      async_g2l_b128(base + (s + 1) * 1024 + tid * 4, &stage[cur ^ 1][tid]);
      wait_async1();          // tile s landed, tile s+1 still in flight
    } else {
      wait_async0();
    }
    v4f v = stage[cur][tid];  // ds_load_b128 of this lane's own 16B
    #pragma unroll
    for (int c = 0; c < 4; ++c) {
      uint32_t key = __float_as_uint(v[c]);
      if ((key & maskHigh) == prefix)
        atomicAdd(&lhist[(key >> shift) & (uint32_t)(nbins - 1)], 1u);  // ds_add
    }
  }
  __syncthreads();

  uint32_t* gh = ws + HIST_OFF + (size_t)b * HIST_BINS;
  for (int i = tid; i < nbins; i += 256) {
    uint32_t c = lhist[i];
    if (c) atomicAdd(&gh[i], c);   // global_atomic_add_u32 (L2)
  }
}

// ---- radix scan: pick digit bin containing the k-th smallest ----------------
__global__ __launch_bounds__(256) void radix_scan_kernel(uint32_t* ws, int shift,
                                                         int nbins) {
  __shared__ uint32_t sh[HIST_BINS];
  const int b = blockIdx.x, tid = threadIdx.x;
  uint32_t* gh = ws + HIST_OFF + (size_t)b * HIST_BINS;
  for (int i = tid; i < nbins; i += 256) sh[i] = gh[i];
  __syncthreads();
  if (tid == 0) {
    uint32_t kRem = ws[STATE_OFF + 2 * b + 1];
    uint32_t cum = 0;
    int j = nbins - 1;
    for (int i = 0; i < nbins; ++i) {
      uint32_t c = sh[i];
      if (cum + c >= kRem) { j = i; break; }
      cum += c;
    }
    ws[STATE_OFF + 2 * b + 0] |= ((uint32_t)j) << shift;
    ws[STATE_OFF + 2 * b + 1] = kRem - cum;   // always >= 1
  }
  __syncthreads();
  for (int i = tid; i < nbins; i += 256) gh[i] = 0u;  // ready for next round
}

// ---- deterministic tie handling --------------------------------------------
// Reference top_k keeps lowest flat index first among equal values; collect
// indices of elements exactly equal to the 32-bit threshold, then keep the
// kRem lowest indices (value-ordered selection => deterministic output even
// though atomic fill order is not).
__global__ __launch_bounds__(256) void tie_collect_kernel(
    const float* __restrict__ cost, uint32_t* __restrict__ ws) {
  const int b = blockIdx.y;
  const uint32_t thresh = ws[STATE_OFF + 2 * b];
  const uint32_t i4 = blockIdx.x * blockDim.x + threadIdx.x;
  const size_t base = (size_t)b * NELEM + (size_t)i4 * 4;
  v4f v = *(const v4f*)(cost + base);
  #pragma unroll
  for (int c = 0; c < 4; ++c) {
    if (__float_as_uint(v[c]) == thresh) {
      uint32_t p = atomicAdd(ws + LISTCNT_OFF + b, 1u);
      if (p < LIST_CAP) ws[LIST_OFF + b * LIST_CAP + p] = i4 * 4 + (uint32_t)c;
    }
  }
}

__global__ void tie_resolve_kernel(uint32_t* ws) {
  const int b = blockIdx.x;
  if (threadIdx.x != 0) return;
  const uint32_t cnt  = ws[LISTCNT_OFF + b];
  const uint32_t kRem = ws[STATE_OFF + 2 * b + 1];
  uint32_t cutoff = 0xFFFFFFFFu;               // keep all ties
  if (cnt > kRem && cnt <= LIST_CAP) {
    const uint32_t* lst = ws + LIST_OFF + b * LIST_CAP;
    uint32_t last = 0u;
    for (uint32_t i = 0; i < kRem; ++i) {      // kRem-th smallest index
      uint32_t mn = 0xFFFFFFFFu;
      for (uint32_t j = 0; j < cnt; ++j) {
        uint32_t x = lst[j];
        if (x < mn && (i == 0 || x > last)) mn = x;
      }
      last = mn;
    }
    cutoff = last;
  }
  ws[CUTOFF_OFF + b] = cutoff;
}

// ---- final masked streaming write ------------------------------------------
__global__ __launch_bounds__(256) void sparse_mask_kernel(
    const float* __restrict__ cost, float* __restrict__ out,
    const uint32_t* __restrict__ ws) {
  const int b = blockIdx.y;
  const uint32_t thresh = ws[STATE_OFF + 2 * b];
  const uint32_t cutoff = ws[CUTOFF_OFF + b];
  const uint32_t i4 = blockIdx.x * blockDim.x + threadIdx.x;
  const size_t base = (size_t)b * NELEM + (size_t)i4 * 4;
  __builtin_prefetch(cost + base + 16384, 0, 0);   // global_prefetch_b8
  v4f v = *(const v4f*)(cost + base);
  v4f r;
  #pragma unroll
  for (int c = 0; c < 4; ++c) {
    uint32_t key = __float_as_uint(v[c]);
    uint32_t idx = i4 * 4 + (uint32_t)c;
    bool keep = (key < thresh) || ((key == thresh) && (idx <= cutoff));
    r[c] = keep ? v[c] : 0.0f;
  }
  __builtin_nontemporal_store(r, (v4f*)(out + base));  // NT b128: write-once
}

// ============================================================================
extern "C" void kernel_launch(void* const* d_in, const int* in_sizes, int n_in,
                              void* d_out, int out_size, void* d_ws,
                              size_t ws_size, hipStream_t stream) {
  const float* src  = (const float*)d_in[0];
  const float* tgt  = (const float*)d_in[1];
  const float* cost = (const float*)d_in[2];
  float* out = (float*)d_out;
  uint32_t* ws = (uint32_t*)d_ws;

  // Outputs 0,1: verbatim copies of source/target.
  (void)hipMemcpyAsync(out, src, (size_t)SRC_ELEMS * sizeof(float),
                       hipMemcpyDeviceToDevice, stream);
  (void)hipMemcpyAsync(out + SRC_ELEMS, tgt, (size_t)SRC_ELEMS * sizeof(float),
                       hipMemcpyDeviceToDevice, stream);

  // Re-initialize workspace every call (harness does not re-poison).
  (void)hipMemsetAsync(d_ws, 0, (size_t)WS_U32 * sizeof(uint32_t), stream);
  init_state_kernel<<<1, 32, 0, stream>>>(ws);

  // 3-round radix select over 32-bit keys: digits 11 + 11 + 10 bits.
  const struct { uint32_t maskHigh; int shift; int nbins; } R[3] = {
      {0x00000000u, 21, 2048},
      {0xFFE00000u, 10, 2048},
      {0xFFFFFC00u,  0, 1024},
  };
  const dim3 histGrid(NELEM / (STAGES * 1024), NBATCH);  // (64, 16)
  for (int r = 0; r < 3; ++r) {
    radix_hist_kernel<<<histGrid, 256, 0, stream>>>(cost, ws, R[r].maskHigh,
                                                    R[r].shift, R[r].nbins);
    radix_scan_kernel<<<NBATCH, 256, 0, stream>>>(ws, R[r].shift, R[r].nbins);
  }

  // Deterministic tie resolution + final masked write.
  const dim3 elemGrid(NELEM / (256 * 4), NBATCH);        // (1024, 16)
  tie_collect_kernel<<<elemGrid, 256, 0, stream>>>(cost, ws);
  tie_resolve_kernel<<<NBATCH, 32, 0, stream>>>(ws);
  sparse_mask_kernel<<<elemGrid, 256, 0, stream>>>(cost, out + 2 * (size_t)SRC_ELEMS,
                                                   ws);
}